// Attn_29274497090270
// MI455X (gfx1250) — compile-verified
//
#include <hip/hip_runtime.h>
#include <math.h>

typedef __attribute__((ext_vector_type(2))) float v2f;
typedef __attribute__((ext_vector_type(4))) float v4f;
typedef __attribute__((ext_vector_type(8))) float v8f;

#define BS 32
#define SEQ 2048
#define HID 1024

// ---------------------------------------------------------------------------
// K1: v[b,h] = sum_k hidden[b,k] * W[k,h]   (M=32, K=1024, N=1024 fp32 GEMM)
// One wave per 16-wide N tile; covers both M tiles (batch = 32 = 2x16).
// Uses V_WMMA_F32_16X16X4_F32; K advances 4 per pair of WMMAs.
// Fragment layouts per CDNA5 ISA 7.12.2 (32-bit A 16x4 / 32-bit C/D 16x16).
// ---------------------------------------------------------------------------
__global__ __launch_bounds__(32) void k1_hidden_times_W(
    const float* __restrict__ hidden,   // [BS, HID]
    const float* __restrict__ W,        // [HID, HID]  (W[k*HID + h])
    float* __restrict__ v)              // [BS, HID]
{
    const int lane  = threadIdx.x;          // 0..31, EXEC all-ones (WMMA req.)
    const int n0    = blockIdx.x * 16;      // N tile base (output h index)
    const int mrow  = lane & 15;            // row within 16-tile / N column
    const int khalf = (lane >> 4) * 2;      // lanes 0-15 -> K 0,1 ; 16-31 -> K 2,3

    v8f acc0 = {};   // M = 0..15
    v8f acc1 = {};   // M = 16..31

    for (int k0 = 0; k0 < HID; k0 += 4) {
        // A fragments: 16x4 slice of hidden (per-lane 2 floats)
        v2f a0, a1, bfrag;
        a0.x = hidden[ mrow       * HID + k0 + khalf    ];
        a0.y = hidden[ mrow       * HID + k0 + khalf + 1];
        a1.x = hidden[(mrow + 16) * HID + k0 + khalf    ];
        a1.y = hidden[(mrow + 16) * HID + k0 + khalf + 1];
        // B fragment: 4x16 slice of W (K rows k0+khalf..+1, N cols n0+mrow)
        bfrag.x = W[(k0 + khalf    ) * HID + n0 + mrow];
        bfrag.y = W[(k0 + khalf + 1) * HID + n0 + mrow];

        acc0 = __builtin_amdgcn_wmma_f32_16x16x4_f32(
                   false, a0, false, bfrag, (short)0, acc0, false, false);
        acc1 = __builtin_amdgcn_wmma_f32_16x16x4_f32(
                   false, a1, false, bfrag, (short)0, acc1, false, false);
    }

    // D layout: VGPR r -> (lanes 0-15: M=r, N=lane) (lanes 16-31: M=r+8, N=lane-16)
    const int ncol  = n0 + (lane & 15);
    const int mbase = (lane >> 4) * 8;
#pragma unroll
    for (int r = 0; r < 8; ++r) {
        v[(mbase + r)      * HID + ncol] = acc0[r];
        v[(mbase + r + 16) * HID + ncol] = acc1[r];
    }
}

// ---------------------------------------------------------------------------
// K2: scores[b,s] = dot(enc[b,s,:], v[b,:])
// HBM-bound: streams 268 MB of enc exactly once (268 MB / 23.3 TB/s ~ 11.5us
// floor). enc is read with NT temporal hint (bigger than the 192 MB L2 and
// touched once -> don't thrash reusable lines); v stays on the cached path
// (128 KB, reused by all 2048 rows of its batch). 8 waves/block, one row per
// wave, float4 loads -> 512 B contiguous per wave-load. Four independent
// accumulator chains so FMA latency never gates the memory pipe.
// ---------------------------------------------------------------------------
__global__ __launch_bounds__(256) void k2_scores(
    const float* __restrict__ enc,      // [BS, SEQ, HID]
    const float* __restrict__ v,        // [BS, HID]
    float* __restrict__ scores)         // [BS*SEQ]
{
    const int  wave = threadIdx.x >> 5;
    const int  lane = threadIdx.x & 31;
    const long row  = (long)blockIdx.x * 8 + wave;   // 0 .. BS*SEQ-1
    const int  b    = (int)(row >> 11);              // row / SEQ

    const v4f* er = (const v4f*)(enc + row * (long)HID) + lane;
    const v4f* vr = (const v4f*)(v   + (long)b * HID)   + lane;

    float a0 = 0.0f, a1 = 0.0f, a2 = 0.0f, a3 = 0.0f;
#pragma unroll
    for (int i = 0; i < HID / 128; ++i) {            // 8 iters * 32 lanes * 4
        v4f e = __builtin_nontemporal_load(er + i * 32);
        v4f w = vr[i * 32];
        a0 = fmaf(e.x, w.x, a0);
        a1 = fmaf(e.y, w.y, a1);
        a2 = fmaf(e.z, w.z, a2);
        a3 = fmaf(e.w, w.w, a3);
    }
    float acc = (a0 + a1) + (a2 + a3);
#pragma unroll
    for (int off = 16; off > 0; off >>= 1)
        acc += __shfl_xor(acc, off, 32);
    if (lane == 0) scores[row] = acc;
}

// ---------------------------------------------------------------------------
// K3: softmax over SEQ per batch row (bias dropped: softmax is invariant to
// the per-row constant shift c[b] = dot(bias, hidden[b])).
// ---------------------------------------------------------------------------
__global__ __launch_bounds__(256) void k3_softmax(
    const float* __restrict__ scores,   // [BS, SEQ]
    float* __restrict__ out)            // [BS, SEQ]
{
    __shared__ float red[8];
    const int b   = blockIdx.x;
    const int tid = threadIdx.x;
    const float* sr = scores + (long)b * SEQ;

    float vals[8];
    float m = -INFINITY;
#pragma unroll
    for (int i = 0; i < 8; ++i) {
        vals[i] = sr[tid + i * 256];
        m = fmaxf(m, vals[i]);
    }
#pragma unroll
    for (int off = 16; off > 0; off >>= 1)
        m = fmaxf(m, __shfl_xor(m, off, 32));
    if ((tid & 31) == 0) red[tid >> 5] = m;
    __syncthreads();
    float mm = red[0];
#pragma unroll
    for (int i = 1; i < 8; ++i) mm = fmaxf(mm, red[i]);

    float sum = 0.0f;
#pragma unroll
    for (int i = 0; i < 8; ++i) {
        vals[i] = expf(vals[i] - mm);
        sum += vals[i];
    }
#pragma unroll
    for (int off = 16; off > 0; off >>= 1)
        sum += __shfl_xor(sum, off, 32);
    __syncthreads();
    if ((tid & 31) == 0) red[tid >> 5] = sum;
    __syncthreads();
    float tot = 0.0f;
#pragma unroll
    for (int i = 0; i < 8; ++i) tot += red[i];
    const float inv = 1.0f / tot;

#pragma unroll
    for (int i = 0; i < 8; ++i)
        out[(long)b * SEQ + tid + i * 256] = vals[i] * inv;
}

// ---------------------------------------------------------------------------
extern "C" void kernel_launch(void* const* d_in, const int* in_sizes, int n_in,
                              void* d_out, int out_size, void* d_ws, size_t ws_size,
                              hipStream_t stream) {
    const float* hidden = (const float*)d_in[0];   // [BS, HID]
    const float* enc    = (const float*)d_in[1];   // [BS, SEQ, HID]
    const float* W      = (const float*)d_in[2];   // [HID, HID]
    // d_in[3] = bias: mathematically dead (softmax shift-invariance)

    float* v      = (float*)d_ws;                  // BS*HID floats (128 KB)
    float* scores = v + BS * HID;                  // BS*SEQ floats (256 KB)

    k1_hidden_times_W<<<HID / 16, 32, 0, stream>>>(hidden, W, v);
    k2_scores<<<(BS * SEQ) / 8, 256, 0, stream>>>(enc, v, scores);
    k3_softmax<<<BS, 256, 0, stream>>>(scores, (float*)d_out);
}